// PraxisAttention_19473381720681
// MI455X (gfx1250) — compile-verified
//
#include <hip/hip_runtime.h>
#include <math.h>

// Problem constants (match reference setup_inputs)
#define Hh    16
#define DHd   64
#define Dd    1024        // H*DH
#define Bb    2
#define Ss    2048
#define MEMN  1000
#define KTT   10          // k_similar + k_contiguous
#define THRESHC 0.5f

typedef float v2f __attribute__((ext_vector_type(2)));
typedef float v8f __attribute__((ext_vector_type(8)));
typedef unsigned int u32;
typedef u32 u32x4 __attribute__((ext_vector_type(4)));
typedef int  i32x4 __attribute__((ext_vector_type(4)));
typedef int  i32x8 __attribute__((ext_vector_type(8)));

// D = A(16x4,f32) * B(4x16,f32) + C(16x16,f32), wave32 CDNA5 WMMA.
static __device__ __forceinline__ v8f wmma4(v2f a, v2f b, v8f c) {
  return __builtin_amdgcn_wmma_f32_16x16x4_f32(false, a, false, b, (short)0, c,
                                               false, false);
}

// ---------------------------------------------------------------------------
// TDM: DMA a 2D f32 tile (tile_x elems wide, tile_y rows, row stride in elems)
// from global memory into LDS at byte offset lds_off. LDS padding: +16 dwords
// after every 64 dwords -> effective LDS row stride 80 floats (bank-friendly).
// D# layout per CDNA5 ISA.
// clang-23 builtin arity: (g0 u32x4, g1 i32x8, g2 i32x4, g3 i32x4, i32x8, cpol)
// ---------------------------------------------------------------------------
static __device__ __forceinline__ void tdm_load_2d_f32(
    u32 lds_off, const float* gaddr, u32 tile_x, u32 tile_y,
    u32 tensor_x, u32 tensor_y, u32 row_stride) {
  const unsigned long long ga = (unsigned long long)(const void*)gaddr;
  u32x4 g0;
  g0.x = 1u;                                       // count=1, user D#
  g0.y = lds_off;                                  // LDS byte address
  g0.z = (u32)(ga & 0xFFFFFFFFull);                // global_addr[31:0]
  g0.w = (u32)((ga >> 32) & 0x1FFFFFFull) | (2u << 30);   // addr[56:32], type=2
  i32x8 g1;
  g1[0] = (int)((2u << 16)                          // data_size = 4B
              | (1u << 20)                          // pad_enable
              | (5u << 22)                          // pad_interval: 64 dwords
              | (15u << 25));                       // pad_amount: 16 dwords
  g1[1] = (int)((tensor_x & 0xFFFFu) << 16);        // tensor_dim0[15:0] @ bit48
  g1[2] = (int)(((tensor_x >> 16) & 0xFFFFu)        // tensor_dim0[31:16]
              | ((tensor_y & 0xFFFFu) << 16));      // tensor_dim1[15:0]
  g1[3] = (int)(((tensor_y >> 16) & 0xFFFFu)        // tensor_dim1[31:16]
              | ((tile_x & 0xFFFFu) << 16));        // tile_dim0
  g1[4] = (int)(tile_y & 0xFFFFu);                  // tile_dim1 (tile_dim2=0)
  g1[5] = (int)row_stride;                          // tensor_dim0_stride[31:0]
  g1[6] = 0;                                        // stride hi, dim1_stride
  g1[7] = 0;
  const i32x4 z4 = {0, 0, 0, 0};                    // 2D: groups 2/3 unused
  const i32x8 z8 = {0, 0, 0, 0, 0, 0, 0, 0};
  __builtin_amdgcn_tensor_load_to_lds(g0, g1, z4, z4, z8, 0);
}

// ---------------------------------------------------------------------------
// Projection GEMM: Y[M,N] = X[M,K] @ W[N,K]^T   (row-major, f32 WMMA)
// One wave computes a 32x64 tile of Y: 2 A-frags x 4 B-frags = 8 WMMAs per
// k-step of 4 -> ~10.7 flops per L2 byte per lane (2x round-1 intensity).
// ---------------------------------------------------------------------------
__global__ __launch_bounds__(256) void proj_gemm_wmma(
    const float* __restrict__ X, const float* __restrict__ W,
    float* __restrict__ Y, int M, int N, int K) {
  const int lane = threadIdx.x & 31;
  const int wave = (int)((blockIdx.x * blockDim.x + threadIdx.x) >> 5);
  const int ngroups = N >> 6;                 // N/64
  const int mt = wave / ngroups;              // 32-row tile index
  const int ng = wave - mt * ngroups;
  if (mt * 32 >= M) return;
  const int r  = lane & 15;
  const int kh = lane >> 4;
  const int kb = 2 * kh;
  const float* x0 = X + (size_t)(mt * 32 + r) * K;
  const float* x1 = x0 + (size_t)16 * K;
  const float* w0 = W + (size_t)(ng * 64 + r) * K;
  const float* w1 = w0 + (size_t)16 * K;
  const float* w2 = w0 + (size_t)32 * K;
  const float* w3 = w0 + (size_t)48 * K;
  v8f c00 = {0,0,0,0,0,0,0,0};
  v8f c01 = c00, c02 = c00, c03 = c00;        // rows [0,16), cols 0/16/32/48
  v8f c10 = c00, c11 = c00, c12 = c00, c13 = c00;  // rows [16,32)
#pragma unroll 4
  for (int kk = 0; kk < K; kk += 4) {
    v2f a0 = *(const v2f*)(x0 + kk + kb);
    v2f a1 = *(const v2f*)(x1 + kk + kb);
    v2f b0 = *(const v2f*)(w0 + kk + kb);
    v2f b1 = *(const v2f*)(w1 + kk + kb);
    v2f b2 = *(const v2f*)(w2 + kk + kb);
    v2f b3 = *(const v2f*)(w3 + kk + kb);
    c00 = wmma4(a0, b0, c00);  c10 = wmma4(a1, b0, c10);
    c01 = wmma4(a0, b1, c01);  c11 = wmma4(a1, b1, c11);
    c02 = wmma4(a0, b2, c02);  c12 = wmma4(a1, b2, c12);
    c03 = wmma4(a0, b3, c03);  c13 = wmma4(a1, b3, c13);
  }
  float* y0 = Y + (size_t)(mt * 32 + 8 * kh) * N + ng * 64 + r;
  float* y1 = y0 + (size_t)16 * N;
#pragma unroll
  for (int rr = 0; rr < 8; ++rr) {
    y0[(size_t)rr * N]      = c00[rr];
    y0[(size_t)rr * N + 16] = c01[rr];
    y0[(size_t)rr * N + 32] = c02[rr];
    y0[(size_t)rr * N + 48] = c03[rr];
    y1[(size_t)rr * N]      = c10[rr];
    y1[(size_t)rr * N + 16] = c11[rr];
    y1[(size_t)rr * N + 32] = c12[rr];
    y1[(size_t)rr * N + 48] = c13[rr];
  }
}

// ---------------------------------------------------------------------------
// Episodic memory scan (sequential over batch, one 256-thread workgroup).
// surprise = ||k_t - k_{t-1}||, thr = mean + 0.5*std(ddof=1), segment means
// -> FIFO (circular; only last MEM valid events readable), top-KT cosine.
// ---------------------------------------------------------------------------
__global__ __launch_bounds__(256) void episodic_scan_kernel(
    const float* __restrict__ kfb, float* __restrict__ memb,
    float* __restrict__ retb) {
  __shared__ float s_sur[Ss];
  __shared__ int   s_ends[Ss];
  __shared__ float s_sims[MEMN];
  __shared__ float rbv[256];
  __shared__ int   rbi[256];
  __shared__ float s_thr, s_qn;
  __shared__ int   s_nseg, s_total, s_bidx;
  const int tid = threadIdx.x, nt = blockDim.x;
  if (tid == 0) s_total = 0;
  __syncthreads();
  for (int b = 0; b < Bb; ++b) {
    const float* kb_ = kfb + (size_t)b * Ss * Dd;
    for (int t = tid; t < Ss; t += nt) {
      float acc = 0.f;
      if (t > 0) {
        const float* x0 = kb_ + (size_t)(t - 1) * Dd;
        const float* x1 = kb_ + (size_t)t * Dd;
        for (int d = 0; d < Dd; ++d) { float df = x1[d] - x0[d]; acc += df * df; }
        acc = sqrtf(acc);
      }
      s_sur[t] = acc;
    }
    __syncthreads();
    float ls = 0.f, lq = 0.f;
    for (int t = tid; t < Ss; t += nt) { float v = s_sur[t]; ls += v; lq += v * v; }
    rbv[tid] = ls; __syncthreads();
    for (int st = 128; st > 0; st >>= 1) { if (tid < st) rbv[tid] += rbv[tid + st]; __syncthreads(); }
    const float tot = rbv[0]; __syncthreads();
    rbv[tid] = lq; __syncthreads();
    for (int st = 128; st > 0; st >>= 1) { if (tid < st) rbv[tid] += rbv[tid + st]; __syncthreads(); }
    const float totq = rbv[0]; __syncthreads();
    if (tid == 0) {
      float mean = tot / (float)Ss;
      float var  = (totq - (float)Ss * mean * mean) / (float)(Ss - 1);
      s_thr = mean + THRESHC * sqrtf(fmaxf(var, 0.f));
    }
    __syncthreads();
    if (tid == 0) {
      int ns = 0; const float thr = s_thr;
      for (int t = 0; t < Ss; ++t)
        if (s_sur[t] > thr || t == Ss - 1) s_ends[ns++] = t;
      s_nseg = ns;
    }
    __syncthreads();
    const int nseg = s_nseg;
    const int base = s_total;
    for (int s = 0; s < nseg; ++s) {
      const int st = (s == 0) ? 0 : (s_ends[s - 1] + 1);
      const int en = s_ends[s];
      const float invc = 1.0f / (float)(en - st + 1);
      float* mrow = memb + (size_t)((base + s) % MEMN) * Dd;
      for (int d = tid; d < Dd; d += nt) {
        float acc = 0.f;
        for (int t = st; t <= en; ++t) acc += kb_[(size_t)t * Dd + d];
        mrow[d] = acc * invc;
      }
    }
    __syncthreads();
    if (tid == 0) s_total = base + nseg;
    __syncthreads();
    const int vcount = (s_total < MEMN) ? s_total : MEMN;
    const float* q = kb_ + (size_t)(Ss - 1) * Dd;
    float lqn = 0.f;
    for (int d = tid; d < Dd; d += nt) { float v = q[d]; lqn += v * v; }
    rbv[tid] = lqn; __syncthreads();
    for (int st = 128; st > 0; st >>= 1) { if (tid < st) rbv[tid] += rbv[tid + st]; __syncthreads(); }
    if (tid == 0) s_qn = fmaxf(sqrtf(rbv[0]), 1e-8f);
    __syncthreads();
    const float qn = s_qn;
    for (int i = tid; i < MEMN; i += nt) {
      float sim = -3.4e38f;
      if (i < vcount) {
        const float* mrow = memb + (size_t)i * Dd;
        float dp = 0.f, nn = 0.f;
        for (int d = 0; d < Dd; ++d) { float mv = mrow[d]; dp += mv * q[d]; nn += mv * mv; }
        sim = dp / (fmaxf(sqrtf(nn), 1e-8f) * qn);
      }
      s_sims[i] = sim;
    }
    __syncthreads();
    for (int kt = 0; kt < KTT; ++kt) {
      float bv = -3.4e38f; int bix = 0;
      for (int i = tid; i < MEMN; i += nt)
        if (s_sims[i] > bv) { bv = s_sims[i]; bix = i; }
      rbv[tid] = bv; rbi[tid] = bix; __syncthreads();
      for (int st = 128; st > 0; st >>= 1) {
        if (tid < st && rbv[tid + st] > rbv[tid]) { rbv[tid] = rbv[tid + st]; rbi[tid] = rbi[tid + st]; }
        __syncthreads();
      }
      if (tid == 0) { s_bidx = rbi[0]; s_sims[rbi[0]] = -3.4e38f; }
      __syncthreads();
      const float* mrow = memb + (size_t)s_bidx * Dd;
      float* rrow = retb + (size_t)(b * KTT + kt) * Dd;
      for (int d = tid; d < Dd; d += nt) rrow[d] = mrow[d];
      __syncthreads();
    }
  }
}

// Row pointer into augmented K/V space: j<KT -> retrieved event, else token j-KT.
static __device__ __forceinline__ const float* aug_row(
    int j, int b, int h, const float* __restrict__ kv, const float* __restrict__ ret) {
  if (j < KTT) return ret + (size_t)(b * KTT + j) * Dd + h * DHd;
  int pk = j - KTT;
  if (pk >= Ss) pk = Ss - 1;                 // OOB columns are masked; clamp addr
  return kv + (size_t)(b * Ss + pk) * Dd + h * DHd;
}

// ---------------------------------------------------------------------------
// Flash attention with ALiBi + causal mask over augmented (memory + real) keys.
// One wave per 16-query tile; f32 WMMA for QK^T and PV; online softmax.
// V tiles for jt>=1 are regular 2D tiles of vf -> TDM double-buffered into LDS
// (tensor_load_to_lds + s_wait_tensorcnt); tile 0 (mixed mem/real) is scalar.
// ---------------------------------------------------------------------------
#define VROW 80   // LDS row stride in floats (64 data + 16 pad via TDM pad)
__global__ __launch_bounds__(128) void flash_attn_wmma(
    const float* __restrict__ qf, const float* __restrict__ kf,
    const float* __restrict__ vf, const float* __restrict__ ret,
    const int* __restrict__ amask, float* __restrict__ out) {
  __shared__ float plds[4][16][18];          // per-wave P tile, padded rows
  __shared__ float vlds[4][2][16 * VROW];    // per-wave double-buffered V tile
  const int lane = threadIdx.x & 31;
  const int lw   = threadIdx.x >> 5;
  const int wid  = (int)((blockIdx.x * blockDim.x + threadIdx.x) >> 5);
  const int qt = wid & 127;                  // S/16 = 128 query tiles
  const int h  = (wid >> 7) & 15;
  const int b  = wid >> 11;
  const int r  = lane & 15;
  const int kh = lane >> 4;
  const int kb = 2 * kh;
  const float slope = exp2f(-0.5f * (float)(h + 1));   // 2^(-8*(h+1)/16)

  const float* qrow = qf + (size_t)(b * Ss + qt * 16 + r) * Dd + h * DHd;
  v2f qa[16];
#pragma unroll
  for (int c = 0; c < 16; ++c) qa[c] = *(const v2f*)(qrow + 4 * c + kb);

  v8f o0 = {0,0,0,0,0,0,0,0}, o1 = o0, o2 = o0, o3 = o0;
  float mi[8], li[8];
#pragma unroll
  for (int rr = 0; rr < 8; ++rr) { mi[rr] = -1e30f; li[rr] = 0.f; }

  const int nkeys  = KTT + qt * 16 + 16;     // causal upper bound for this tile
  const int ntiles = (nkeys + 15) >> 4;

  // Prefetch V tile jt=1 (rows pk0..pk0+15 of vf, all real tokens).
  if (ntiles > 1) {
    const int pk0 = 16 - KTT;
    tdm_load_2d_f32((u32)(unsigned long long)(const void*)&vlds[lw][0][0],
                    vf + (size_t)(b * Ss + pk0) * Dd + h * DHd,
                    64u, 16u, 64u, (u32)min(16, Ss - pk0), (u32)Dd);
  }

  for (int jt = 0; jt < ntiles; ++jt) {
    const int  jc      = jt * 16 + r;        // this lane's key column
    const bool isMem   = jc < KTT;
    const bool inRange = jc < KTT + Ss;
    const int  pk      = jc - KTT;
    const float* krow  = aug_row(jc, b, h, kf, ret);
    int mok = 1;
    if (!isMem && inRange) mok = amask[b * Ss + pk];

    // S tile = Q(16x64) . K^T(64x16)
    v8f sa = {0,0,0,0,0,0,0,0};
#pragma unroll
    for (int c = 0; c < 16; ++c) {
      v2f bf = *(const v2f*)(krow + 4 * c + kb);
      sa = wmma4(qa[c], bf, sa);
    }

    // scale + ALiBi + mask + online softmax (row = rr + 8*kh, col = r)
    float p[8], cf[8];
#pragma unroll
    for (int rr = 0; rr < 8; ++rr) {
      const int pq = qt * 16 + rr + 8 * kh;
      float s = sa[rr] * 0.125f;             // 1/sqrt(64)
      if (!isMem) s -= slope * (float)(pq - pk);
      const bool vis = inRange && (mok != 0) && (isMem || (pk <= pq));
      s = vis ? s : -1e30f;
      float v = s;
      v = fmaxf(v, __shfl_xor(v, 1));
      v = fmaxf(v, __shfl_xor(v, 2));
      v = fmaxf(v, __shfl_xor(v, 4));
      v = fmaxf(v, __shfl_xor(v, 8));
      const float nm = fmaxf(mi[rr], v);
      const float c_ = __expf(mi[rr] - nm);
      const float pe = __expf(s - nm);
      float rs = pe;
      rs += __shfl_xor(rs, 1);
      rs += __shfl_xor(rs, 2);
      rs += __shfl_xor(rs, 4);
      rs += __shfl_xor(rs, 8);
      li[rr] = li[rr] * c_ + rs;
      mi[rr] = nm;
      cf[rr] = c_;
      p[rr]  = pe;
    }
#pragma unroll
    for (int rr = 0; rr < 8; ++rr) {
      o0[rr] *= cf[rr]; o1[rr] *= cf[rr]; o2[rr] *= cf[rr]; o3[rr] *= cf[rr];
    }

    // C-layout -> A-layout via per-wave LDS tile (DS ops are in-order per wave)
    __builtin_amdgcn_wave_barrier();
#pragma unroll
    for (int rr = 0; rr < 8; ++rr) plds[lw][rr + 8 * kh][r] = p[rr];
    __builtin_amdgcn_wave_barrier();

    // Issue TDM prefetch for tile jt+1 (double buffer), then wait for current.
    if (jt >= 1) {
      if (jt + 1 < ntiles) {
        const int pk0 = (jt + 1) * 16 - KTT;
        tdm_load_2d_f32((u32)(unsigned long long)(const void*)&vlds[lw][jt & 1][0],
                        vf + (size_t)(b * Ss + pk0) * Dd + h * DHd,
                        64u, 16u, 64u, (u32)min(16, Ss - pk0), (u32)Dd);
        __builtin_amdgcn_s_wait_tensorcnt(1);
      } else {
        __builtin_amdgcn_s_wait_tensorcnt(0);
      }
    }

    // O(16x64) += P(16x16) . V(16x64)
    const float* vbuf = &vlds[lw][(jt - 1) & 1][0];
#pragma unroll
    for (int kc = 0; kc < 4; ++kc) {
      const v2f pa = *(const v2f*)&plds[lw][r][kc * 4 + kb];
      const int k0 = kc * 4 + kb;
      v2f vb0, vb1, vb2, vb3;
      if (jt == 0) {
        const float* v0r = aug_row(k0,     b, h, vf, ret);
        const float* v1r = aug_row(k0 + 1, b, h, vf, ret);
        vb0.x = v0r[r];      vb0.y = v1r[r];
        vb1.x = v0r[16 + r]; vb1.y = v1r[16 + r];
        vb2.x = v0r[32 + r]; vb2.y = v1r[32 + r];
        vb3.x = v0r[48 + r]; vb3.y = v1r[48 + r];
      } else {
        const float* r0 = vbuf + k0 * VROW;
        const float* r1 = vbuf + (k0 + 1) * VROW;
        vb0.x = r0[r];      vb0.y = r1[r];
        vb1.x = r0[16 + r]; vb1.y = r1[16 + r];
        vb2.x = r0[32 + r]; vb2.y = r1[32 + r];
        vb3.x = r0[48 + r]; vb3.y = r1[48 + r];
      }
      o0 = wmma4(pa, vb0, o0);
      o1 = wmma4(pa, vb1, o1);
      o2 = wmma4(pa, vb2, o2);
      o3 = wmma4(pa, vb3, o3);
    }
  }

  // Write normalized output in [B,S,D] token-major layout (pre-Wo)
  float* ob = out + (size_t)(b * Ss + qt * 16 + 8 * kh) * Dd + h * DHd + r;
#pragma unroll
  for (int rr = 0; rr < 8; ++rr) {
    const float inv = 1.0f / li[rr];
    ob[(size_t)rr * Dd]      = o0[rr] * inv;
    ob[(size_t)rr * Dd + 16] = o1[rr] * inv;
    ob[(size_t)rr * Dd + 32] = o2[rr] * inv;
    ob[(size_t)rr * Dd + 48] = o3[rr] * inv;
  }
}

// ---------------------------------------------------------------------------
extern "C" void kernel_launch(void* const* d_in, const int* in_sizes, int n_in,
                              void* d_out, int out_size, void* d_ws, size_t ws_size,
                              hipStream_t stream) {
  const float* x     = (const float*)d_in[0];
  const int*   amask = (const int*)d_in[1];
  const float* Wq    = (const float*)d_in[2];
  const float* Wk    = (const float*)d_in[3];
  const float* Wv    = (const float*)d_in[4];
  const float* Wo    = (const float*)d_in[5];
  float* out = (float*)d_out;
  float* ws  = (float*)d_ws;

  const size_t nBSD = (size_t)Bb * Ss * Dd;      // 4,194,304 floats
  float* qfb  = ws;
  float* kfb  = qfb + nBSD;
  float* vfb  = kfb + nBSD;
  float* atb  = vfb + nBSD;
  float* memb = atb + nBSD;                      // MEM x D event FIFO
  float* retb = memb + (size_t)MEMN * Dd;        // B x KT x D retrieved

  const int M = Bb * Ss, N = Dd, K = Dd;
  const dim3 gb(256), tb(256);                   // 2048 waves = (M/32)*(N/64)
  proj_gemm_wmma<<<gb, tb, 0, stream>>>(x, Wq, qfb, M, N, K);
  proj_gemm_wmma<<<gb, tb, 0, stream>>>(x, Wk, kfb, M, N, K);
  proj_gemm_wmma<<<gb, tb, 0, stream>>>(x, Wv, vfb, M, N, K);
  episodic_scan_kernel<<<1, 256, 0, stream>>>(kfb, memb, retb);
  flash_attn_wmma<<<1024, 128, 0, stream>>>(qfb, kfb, vfb, retb, amask, atb);
  proj_gemm_wmma<<<gb, tb, 0, stream>>>(atb, Wo, out, M, N, K);
  (void)in_sizes; (void)n_in; (void)out_size; (void)ws_size;
}